// SequenceAttLayer_40793599377880
// MI455X (gfx1250) — compile-verified
//
#include <hip/hip_runtime.h>
#include <math.h>

typedef __attribute__((ext_vector_type(2))) float v2f;
typedef __attribute__((ext_vector_type(4))) float v4f;
typedef __attribute__((ext_vector_type(8))) float v8f;

#define BATCH 8192
#define NQ    5
#define TLEN  200
#define EMB   64
#define TPAD  208      // T padded to multiple of 16
#define KSTR  68       // keys LDS row stride (floats): mult of 4 (16B), 68%64=4 spreads banks
#define QSTR  68       // q2 LDS row stride
#define PSTR  212      // probs LDS row stride: mult of 4, 212%64=20 spreads banks

// ---- CDNA5 async global->LDS path (guarded; falls back to vmem+ds) --------
#if __has_builtin(__builtin_amdgcn_global_load_async_to_lds_b128)
#define HAVE_ASYNC_LDS 1
typedef int v4i_raw __attribute__((vector_size(16)));              // matches builtin param
typedef __attribute__((address_space(1))) v4i_raw as1_v4i;         // global
typedef __attribute__((address_space(3))) v4i_raw as3_v4i;         // LDS
#endif

__device__ __forceinline__ void copy_b128_to_lds(const float* __restrict__ g,
                                                 float* l) {
#if defined(HAVE_ASYNC_LDS)
    __builtin_amdgcn_global_load_async_to_lds_b128((as1_v4i*)g, (as3_v4i*)l, 0, 0);
#else
    *(v4f*)l = *(const v4f*)g;
#endif
}

__device__ __forceinline__ void drain_async_lds() {
#if defined(HAVE_ASYNC_LDS)
#if __has_builtin(__builtin_amdgcn_s_wait_asynccnt)
    __builtin_amdgcn_s_wait_asynccnt(0);
#else
    asm volatile("s_wait_asynccnt 0x0" ::: "memory");
#endif
#endif
}

// ---------------------------------------------------------------------------
// Kernel 1: fold projections.  C[e1][e2] = sum_o W1[o][e1] * W2[o][e2] = (W1^T W2)
// (scores = (q W1^T)(K W2^T)^T = q (W1^T W2) K^T, so the 200x64x64 keys
//  projection folds away entirely -> kernel becomes memory-bound on keys.)
// ---------------------------------------------------------------------------
__global__ void fold_weights_kernel(const float* __restrict__ W1,
                                    const float* __restrict__ W2,
                                    float* __restrict__ C) {
    int i = blockIdx.x * blockDim.x + threadIdx.x;
    if (i >= EMB * EMB) return;
    int e1 = i >> 6, e2 = i & 63;
    float s = 0.f;
    #pragma unroll 8
    for (int o = 0; o < EMB; ++o)
        s = fmaf(W1[o * EMB + e1], W2[o * EMB + e2], s);
    C[i] = s;
}

// ---------------------------------------------------------------------------
// Kernel 2: per-batch attention. One 256-thread block (8 wave32) per batch.
//   q2     = q @ C                       [5,64]   (tiny, plain VALU)
//   scores = q2 @ K^T  (WMMA f32 16x16x4), mask->0, * 1/8
//   probs  = softmax over t<200
//   out    = probs @ K (WMMA f32 16x16x4)
// ---------------------------------------------------------------------------
__global__ __launch_bounds__(256)
void seq_att_kernel(const float* __restrict__ queries,
                    const float* __restrict__ keys,
                    const int*   __restrict__ keys_length,
                    const float* __restrict__ C,
                    float* __restrict__ out) {
    __shared__ float kLDS[TPAD * KSTR];   // 56,576 B : keys tile (rows 200..207 zeroed)
    __shared__ float q2LDS[16 * QSTR];    //  4,352 B : projected queries, rows 5..15 zero
    __shared__ float pLDS[16 * PSTR];     // 13,568 B : scores -> probs (rows 0..4 live)
    __shared__ float cLDS[EMB * EMB];     // 16,384 B : folded weight matrix
    __shared__ float qIn[NQ * EMB];       //  1,280 B : raw queries

    const int b    = blockIdx.x;
    const int tid  = threadIdx.x;
    const int lane = tid & 31;
    const int wave = tid >> 5;
    const int lo   = lane & 15;     // row/col index within 16
    const int hi   = lane >> 4;     // half-wave select (K pair offset)

    // ---- stage keys tile into LDS via async DMA (ASYNCcnt path) ----
    const float* kg = keys + (size_t)b * TLEN * EMB;
    for (int i = tid; i < TLEN * (EMB / 4); i += 256) {
        int t = i >> 4;             // 16 float4 per row
        int c = (i & 15) << 2;
        copy_b128_to_lds(kg + t * EMB + c, &kLDS[t * KSTR + c]);
    }
    // ---- stage folded weights (L2-resident) and queries ----
    for (int i = tid; i < EMB * EMB / 4; i += 256)
        copy_b128_to_lds(C + i * 4, &cLDS[i * 4]);
    const float* qg = queries + (size_t)b * NQ * EMB;
    for (int i = tid; i < NQ * EMB / 4; i += 256)
        copy_b128_to_lds(qg + i * 4, &qIn[i * 4]);

    // ---- zero pads with plain LDS stores (overlap with in-flight DMA) ----
    for (int i = tid; i < 8 * KSTR; i += 256)          // keys rows 200..207
        kLDS[TLEN * KSTR + i] = 0.f;
    for (int i = tid; i < (16 - NQ) * QSTR; i += 256)  // q2 pad rows 5..15
        q2LDS[NQ * QSTR + i] = 0.f;

    drain_async_lds();
    __syncthreads();

    // ---- q2 = qIn @ C  (320 dot products of length 64; negligible) ----
    for (int i = tid; i < NQ * EMB; i += 256) {
        int m = i >> 6, e = i & 63;
        float s = 0.f;
        #pragma unroll 8
        for (int k = 0; k < EMB; ++k)
            s = fmaf(qIn[m * EMB + k], cLDS[k * EMB + e], s);
        q2LDS[m * QSTR + e] = s;
    }
    __syncthreads();

    const int   thre  = TLEN - keys_length[b];   // mask: t < thre -> score 0
    const float scale = 0.125f;                  // 1/sqrt(64)

    // ---- scores[16, 208] = q2[16,64] @ K^T : 13 N-tiles over 8 waves ----
    for (int nt = wave; nt < TPAD / 16; nt += 8) {
        const int n0 = nt * 16;
        v8f acc = {};
        #pragma unroll
        for (int ks = 0; ks < EMB / 4; ++ks) {
            const int k0 = ks * 4 + 2 * hi;
            // A: lane -> row lo of q2, K pair (k0, k0+1)
            v2f a = *(const v2f*)(&q2LDS[lo * QSTR + k0]);
            // B[k][n] = K[n0+n][k]: lane -> col lo, contiguous K pair in keys row
            v2f bb = *(const v2f*)(&kLDS[(n0 + lo) * KSTR + k0]);
            acc = __builtin_amdgcn_wmma_f32_16x16x4_f32(
                      false, a, false, bb, (short)0, acc, false, false);
        }
        if (hi == 0) {                    // D: lanes 0..15 hold M=r (r<8); need rows 0..4
            const int t = n0 + lo;
            #pragma unroll
            for (int r = 0; r < NQ; ++r) {
                float v = (t < thre || t >= TLEN) ? 0.f : acc[r] * scale;
                pLDS[r * PSTR + t] = v;   // pad cols [200,208) get 0 here (tile 12)
            }
        }
    }
    __syncthreads();

    // ---- softmax over t in [0,200): one wave per query row ----
    if (wave < NQ) {
        float* row = &pLDS[wave * PSTR];
        float mx = -INFINITY;
        for (int t = lane; t < TLEN; t += 32) mx = fmaxf(mx, row[t]);
        #pragma unroll
        for (int off = 16; off; off >>= 1) mx = fmaxf(mx, __shfl_xor(mx, off, 32));
        float sum = 0.f;
        for (int t = lane; t < TLEN; t += 32) {
            float e = __expf(row[t] - mx);
            row[t] = e;
            sum += e;
        }
        #pragma unroll
        for (int off = 16; off; off >>= 1) sum += __shfl_xor(sum, off, 32);
        const float inv = 1.f / sum;
        for (int t = lane; t < TLEN; t += 32) row[t] *= inv;
        for (int t = TLEN + lane; t < TPAD; t += 32) row[t] = 0.f;  // K-pad for out GEMM
    }
    __syncthreads();

    // ---- out[16,64] = probs[16,208] @ K[208,64] : 4 N-tiles, waves 0..3 ----
    if (wave < 4) {
        const int n0 = wave * 16;
        v8f acc = {};
        for (int ks = 0; ks < TPAD / 4; ++ks) {   // 52 WMMA steps
            const int k0 = ks * 4 + 2 * hi;
            // A: lane -> row lo of probs, contiguous K pair
            v2f a = *(const v2f*)(&pLDS[lo * PSTR + k0]);
            // B[k][n] = K[k][n0+n]: two B32 LDS reads (row stride KSTR)
            v2f bb;
            bb.x = kLDS[k0 * KSTR + n0 + lo];
            bb.y = kLDS[(k0 + 1) * KSTR + n0 + lo];
            acc = __builtin_amdgcn_wmma_f32_16x16x4_f32(
                      false, a, false, bb, (short)0, acc, false, false);
        }
        if (hi == 0) {
            float* og = out + (size_t)b * NQ * EMB;
            #pragma unroll
            for (int r = 0; r < NQ; ++r)
                og[r * EMB + n0 + lo] = acc[r];
        }
    }
}

// ---------------------------------------------------------------------------
extern "C" void kernel_launch(void* const* d_in, const int* in_sizes, int n_in,
                              void* d_out, int out_size, void* d_ws, size_t ws_size,
                              hipStream_t stream) {
    const float* queries     = (const float*)d_in[0];  // [8192,5,64]
    const float* keys        = (const float*)d_in[1];  // [8192,200,64]
    const int*   keys_length = (const int*)  d_in[2];  // [8192]
    // d_in[3] = mask_mat (arange) unused: mask is just t < T - keys_length
    const float* W1          = (const float*)d_in[4];  // [64,64]
    const float* W2          = (const float*)d_in[5];  // [64,64]
    float*       out         = (float*)d_out;          // [8192,5,64]
    float*       C           = (float*)d_ws;           // 64*64 folded weights (16 KB)

    fold_weights_kernel<<<(EMB * EMB + 255) / 256, 256, 0, stream>>>(W1, W2, C);
    seq_att_kernel<<<BATCH, 256, 0, stream>>>(queries, keys, keys_length, C, out);
}